// TripletAttention_11793980194929
// MI455X (gfx1250) — compile-verified
//
#include <hip/hip_runtime.h>
#include <math.h>

typedef __attribute__((ext_vector_type(2))) float v2f;
typedef __attribute__((ext_vector_type(8))) float v8f;

#define EPS 1e-5f
#define NEG_INF (-3.402823466e38f)

// Async global->LDS copy helpers (CDNA5 GLOBAL_LOAD_ASYNC_TO_LDS_*, ASYNCcnt).
// GVS addressing: mem_addr = SGPR base + VGPR 32-bit offset; VDST VGPR holds the
// LDS byte address (low 32 bits of the generic shared pointer).
__device__ __forceinline__ void async_copy_b64(unsigned lds_byte, unsigned goff_byte,
                                               const void* gbase) {
    asm volatile("global_load_async_to_lds_b64 %0, %1, %2"
                 :: "v"(lds_byte), "v"(goff_byte), "s"((unsigned long long)gbase)
                 : "memory");
}
__device__ __forceinline__ void async_copy_b128(unsigned lds_byte, unsigned goff_byte,
                                                const void* gbase) {
    asm volatile("global_load_async_to_lds_b128 %0, %1, %2"
                 :: "v"(lds_byte), "v"(goff_byte), "s"((unsigned long long)gbase)
                 : "memory");
}
__device__ __forceinline__ void async_wait0() {
    asm volatile("s_wait_asynccnt 0" ::: "memory");
}
__device__ __forceinline__ unsigned lds_addr32(const void* p) {
    return (unsigned)(unsigned long long)p;   // flat LDS addr = low 32 bits
}

// ---------------------------------------------------------------------------
// Branch 3: grouped convs (1025 groups, 2 in-ch each), c1 5x5 pad2, c2 3x3 pad1,
// fused BN+ReLU, writes concat tensor xc[B][2050][32][32].
// One block per (group, batch). Input channels 2g,2g+1 are shared by both convs,
// staged once in a zero-padded 36x36 LDS tile via async global->LDS B64 copies.
// ---------------------------------------------------------------------------
__global__ __launch_bounds__(256) void k_conv_branch3(
    const float* __restrict__ x,
    const float* __restrict__ c1w, const float* __restrict__ c1g, const float* __restrict__ c1b,
    const float* __restrict__ c1m, const float* __restrict__ c1v,
    const float* __restrict__ c2w, const float* __restrict__ c2g, const float* __restrict__ c2b,
    const float* __restrict__ c2m, const float* __restrict__ c2v,
    float* __restrict__ xc)
{
    __shared__ float tile[2 * 36 * 36];
    __shared__ float w1s[50];
    __shared__ float w2s[18];
    const int g = blockIdx.x, b = blockIdx.y;
    const int tid = threadIdx.x;
    const float* xin = x + ((size_t)b * 2050 + 2 * g) * 1024;

    // zero-fill the padded tile (covers the halo), stage weights
    for (int i = tid; i < 2 * 36 * 36; i += 256) tile[i] = 0.f;
    if (tid < 50) w1s[tid] = c1w[g * 50 + tid];
    else if (tid >= 64 && tid < 82) w2s[tid - 64] = c2w[g * 18 + (tid - 64)];
    __syncthreads();

    // async-load the 2x32x32 interior directly into LDS (B64 = 2 floats/transfer)
    const unsigned tbase = lds_addr32(tile);
    for (int i = 0; i < 4; ++i) {
        const int t = tid + 256 * i;          // 1024 qword transfers
        const int chn = t >> 9;               // 0..1
        const int rem = t & 511;
        const int r = rem >> 4;               // row 0..31
        const int q = rem & 15;               // qword in row
        const unsigned goff = (unsigned)((chn * 1024 + r * 32 + q * 2) * 4);
        const unsigned loff = tbase + (unsigned)((chn * 1296 + (r + 2) * 36 + 2 + q * 2) * 4);
        async_copy_b64(loff, goff, xin);
    }
    async_wait0();
    __syncthreads();

    const float inv1 = c1g[g] * rsqrtf(c1v[g] + EPS);
    const float add1 = c1b[g] - c1m[g] * inv1;
    const float inv2 = c2g[g] * rsqrtf(c2v[g] + EPS);
    const float add2 = c2b[g] - c2m[g] * inv2;
    float* o1 = xc + ((size_t)b * 2050 + g) * 1024;
    float* o2 = xc + ((size_t)b * 2050 + 1025 + g) * 1024;

    for (int j = 0; j < 4; ++j) {
        const int p = tid + 256 * j;
        const int h = p >> 5, w = p & 31;
        float a1 = 0.f, a2 = 0.f;
#pragma unroll
        for (int ic = 0; ic < 2; ++ic) {
            const float* t = &tile[ic * 1296];
#pragma unroll
            for (int kh = 0; kh < 5; ++kh)
#pragma unroll
                for (int kw = 0; kw < 5; ++kw)
                    a1 += t[(h + kh) * 36 + (w + kw)] * w1s[ic * 25 + kh * 5 + kw];
#pragma unroll
            for (int kh = 0; kh < 3; ++kh)
#pragma unroll
                for (int kw = 0; kw < 3; ++kw)
                    a2 += t[(h + 1 + kh) * 36 + (w + 1 + kw)] * w2s[ic * 9 + kh * 3 + kw];
        }
        o1[p] = fmaxf(a1 * inv1 + add1, 0.f);
        o2[p] = fmaxf(a2 * inv2 + add2, 0.f);
    }
}

// ---------------------------------------------------------------------------
// ZPool for branches 1 & 2 in a single pass over x.
// z1[b][0/1][c][w] = max/mean over h;  z2[b][0/1][h][c] = max/mean over w.
// 32x32 slice staged via one async B128 global->LDS transfer per thread.
// ---------------------------------------------------------------------------
__global__ __launch_bounds__(256) void k_zpool12(
    const float* __restrict__ x, float* __restrict__ z1, float* __restrict__ z2)
{
    __shared__ float sl[1024];
    const int c = blockIdx.x, b = blockIdx.y;
    const int tid = threadIdx.x;
    const float* xin = x + ((size_t)b * 2050 + c) * 1024;

    async_copy_b128(lds_addr32(sl) + (unsigned)(tid * 16), (unsigned)(tid * 16), xin);
    async_wait0();
    __syncthreads();

    if (tid < 32) {
        float mx = NEG_INF, sm = 0.f;
        for (int h = 0; h < 32; ++h) { float v = sl[h * 32 + tid]; mx = fmaxf(mx, v); sm += v; }
        z1[(((size_t)b * 2 + 0) * 2050 + c) * 32 + tid] = mx;
        z1[(((size_t)b * 2 + 1) * 2050 + c) * 32 + tid] = sm * (1.f / 32.f);
    } else if (tid < 64) {
        const int h = tid - 32;
        float mx = NEG_INF, sm = 0.f;
        for (int w = 0; w < 32; ++w) { float v = sl[h * 32 + w]; mx = fmaxf(mx, v); sm += v; }
        z2[(((size_t)b * 2 + 0) * 32 + h) * 2050 + c] = mx;
        z2[(((size_t)b * 2 + 1) * 32 + h) * 2050 + c] = sm * (1.f / 32.f);
    }
}

// ---------------------------------------------------------------------------
// Branch 3 ZPool over 2050 channels of xc. The mean is computed on the matrix
// pipe: partial_sum = ones(16x4) x B(4x16) accumulated via V_WMMA_F32_16X16X4_F32
// (A = all-ones => every D row holds the column sums). Max stays on VALU with a
// cross-half-wave shuffle merge (lanes n and n+16 cover disjoint channel halves).
// Channels are tiled into 16 chunks of 132 (zero-/(-inf)-masked past 2050).
// ---------------------------------------------------------------------------
__global__ __launch_bounds__(256) void k_reduce3_part(
    const float* __restrict__ xc, float* __restrict__ pmax, float* __restrict__ psum)
{
    const int chunk = blockIdx.x, b = blockIdx.y;
    const int tid = threadIdx.x;
    const int lane = tid & 31, wave = tid >> 5;
    const int n = lane & 15;
    const int khalf = lane >> 4;
    const int c0 = chunk * 132;
    const v2f ones = {1.f, 1.f};
    for (int s = 0; s < 8; ++s) {
        const int p = (wave * 8 + s) * 16 + n;
        v8f acc = {0.f, 0.f, 0.f, 0.f, 0.f, 0.f, 0.f, 0.f};
        float vmax = NEG_INF;
        for (int k4 = 0; k4 < 132; k4 += 4) {
            const int cA = c0 + k4 + khalf * 2;   // B VGPR0: K = 0/2 halves
            const int cB = cA + 1;                // B VGPR1: K = 1/3 halves
            const int cAc = cA < 2049 ? cA : 2049;
            const int cBc = cB < 2049 ? cB : 2049;
            const float rA = xc[((size_t)b * 2050 + cAc) * 1024 + p];
            const float rB = xc[((size_t)b * 2050 + cBc) * 1024 + p];
            vmax = fmaxf(vmax, fmaxf((cA < 2050) ? rA : NEG_INF,
                                     (cB < 2050) ? rB : NEG_INF));
            v2f bm;
            bm.x = (cA < 2050) ? rA : 0.f;
            bm.y = (cB < 2050) ? rB : 0.f;
            acc = __builtin_amdgcn_wmma_f32_16x16x4_f32(
                false, ones, false, bm, (short)0, acc, false, false);
        }
        vmax = fmaxf(vmax, __shfl_xor(vmax, 16, 32));
        if (lane < 16) {
            const size_t o = ((size_t)b * 16 + chunk) * 1024 + p;
            pmax[o] = vmax;
            psum[o] = acc[0];
        }
    }
}

__global__ __launch_bounds__(256) void k_reduce3_final(
    const float* __restrict__ pmax, const float* __restrict__ psum, float* __restrict__ z3)
{
    const int b = blockIdx.x, tid = threadIdx.x;
    for (int j = 0; j < 4; ++j) {
        const int p = tid + 256 * j;
        float mx = NEG_INF, sm = 0.f;
        for (int ch = 0; ch < 16; ++ch) {
            const size_t o = ((size_t)b * 16 + ch) * 1024 + p;
            mx = fmaxf(mx, pmax[o]);
            sm += psum[o];
        }
        z3[((size_t)b * 2 + 0) * 1024 + p] = mx;
        z3[((size_t)b * 2 + 1) * 1024 + p] = sm * (1.f / 2050.f);
    }
}

// ---------------------------------------------------------------------------
// Depthwise 7x7 (groups=2) + BN + ReLU, generic over spatial (Hs, Ws) and gate.
// Maps fit comfortably in L2 (<=16.8 MB), so halo re-reads come from cache.
// ---------------------------------------------------------------------------
__global__ __launch_bounds__(256) void k_dwconv7(
    const float* __restrict__ src, float* __restrict__ dst,
    const float* __restrict__ dww, const float* __restrict__ gam, const float* __restrict__ bet,
    const float* __restrict__ mea, const float* __restrict__ var,
    int gi, int Hs, int Ws)
{
    const int bc = blockIdx.y;            // b*2 + ch
    const int ch = bc & 1;
    const int p = blockIdx.x * 256 + threadIdx.x;
    const int HW = Hs * Ws;
    if (p >= HW) return;
    const int hh = p / Ws, ww = p - hh * Ws;
    const float* sp = src + (size_t)bc * HW;
    const float* wt = dww + (gi * 2 + ch) * 49;
    float acc = 0.f;
#pragma unroll
    for (int kh = 0; kh < 7; ++kh) {
        const int y = hh + kh - 3;
        if (y < 0 || y >= Hs) continue;
#pragma unroll
        for (int kw = 0; kw < 7; ++kw) {
            const int xw = ww + kw - 3;
            if (xw < 0 || xw >= Ws) continue;
            acc += sp[y * Ws + xw] * wt[kh * 7 + kw];
        }
    }
    const float inv = gam[gi * 2 + ch] * rsqrtf(var[gi * 2 + ch] + EPS);
    dst[(size_t)bc * HW + p] = fmaxf(acc * inv + (bet[gi * 2 + ch] - mea[gi * 2 + ch] * inv), 0.f);
}

// 1x1 (2->1 ch) + BN + sigmoid
__global__ __launch_bounds__(256) void k_pwconv(
    const float* __restrict__ src, float* __restrict__ dst,
    const float* __restrict__ pww, const float* __restrict__ gam, const float* __restrict__ bet,
    const float* __restrict__ mea, const float* __restrict__ var,
    int gi, int HW)
{
    const size_t idx = (size_t)blockIdx.x * 256 + threadIdx.x;
    if (idx >= (size_t)16 * HW) return;
    const int b = (int)(idx / HW);
    const int p = (int)(idx - (size_t)b * HW);
    const float a = src[((size_t)b * 2 + 0) * HW + p];
    const float c = src[((size_t)b * 2 + 1) * HW + p];
    float y = pww[gi * 2 + 0] * a + pww[gi * 2 + 1] * c;
    const float inv = gam[gi] * rsqrtf(var[gi] + EPS);
    y = y * inv + (bet[gi] - mea[gi] * inv);
    dst[idx] = 1.f / (1.f + __expf(-y));
}

// out[b,c,h,w] = (xc*s2[b,h,w] + x*(s0[b,c,w] + s1[b,h,c])) / 3
__global__ __launch_bounds__(256) void k_final(
    const float* __restrict__ x, const float* __restrict__ xc,
    const float* __restrict__ s0, const float* __restrict__ s1, const float* __restrict__ s2,
    float* __restrict__ out)
{
    const size_t idx = (size_t)blockIdx.x * 256 + threadIdx.x;
    const size_t total = (size_t)16 * 2050 * 1024;
    if (idx >= total) return;
    const int w = (int)(idx & 31);
    const int h = (int)((idx >> 5) & 31);
    const size_t t = idx >> 10;
    const int c = (int)(t % 2050);
    const int b = (int)(t / 2050);
    const float sv2 = s2[(size_t)b * 1024 + h * 32 + w];
    const float sv0 = s0[((size_t)b * 2050 + c) * 32 + w];
    const float sv1 = s1[((size_t)b * 32 + h) * 2050 + c];
    out[idx] = (xc[idx] * sv2 + x[idx] * (sv0 + sv1)) * (1.f / 3.f);
}

// ---------------------------------------------------------------------------
extern "C" void kernel_launch(void* const* d_in, const int* in_sizes, int n_in,
                              void* d_out, int out_size, void* d_ws, size_t ws_size,
                              hipStream_t stream)
{
    const float* x   = (const float*)d_in[0];
    const float* dww = (const float*)d_in[1];
    const float* dwg = (const float*)d_in[2];
    const float* dwb = (const float*)d_in[3];
    const float* dwm = (const float*)d_in[4];
    const float* dwv = (const float*)d_in[5];
    const float* pww = (const float*)d_in[6];
    const float* pwg = (const float*)d_in[7];
    const float* pwb = (const float*)d_in[8];
    const float* pwm = (const float*)d_in[9];
    const float* pwv = (const float*)d_in[10];
    const float* c1w = (const float*)d_in[11];
    const float* c1g = (const float*)d_in[12];
    const float* c1b = (const float*)d_in[13];
    const float* c1m = (const float*)d_in[14];
    const float* c1v = (const float*)d_in[15];
    const float* c2w = (const float*)d_in[16];
    const float* c2g = (const float*)d_in[17];
    const float* c2b = (const float*)d_in[18];
    const float* c2m = (const float*)d_in[19];
    const float* c2v = (const float*)d_in[20];
    float* out = (float*)d_out;

    // workspace carve (floats)
    const size_t XC = (size_t)16 * 2050 * 1024;   // concat conv output
    const size_t Z1 = (size_t)16 * 2 * 2050 * 32; // gate maps (2 ch)
    const size_t S0 = (size_t)16 * 2050 * 32;     // sigmoid maps
    const size_t Z3 = (size_t)16 * 2 * 1024;
    const size_t S2 = (size_t)16 * 1024;
    const size_t PT = (size_t)16 * 16 * 1024;     // channel-chunk partials
    float* wsf  = (float*)d_ws;
    float* xc   = wsf;
    float* z1a  = xc + XC;
    float* z1b  = z1a + Z1;
    float* s0   = z1b + Z1;
    float* z2a  = s0 + S0;
    float* z2b  = z2a + Z1;
    float* s1   = z2b + Z1;
    float* z3a  = s1 + S0;
    float* z3b  = z3a + Z3;
    float* s2   = z3b + Z3;
    float* pmax = s2 + S2;
    float* psum = pmax + PT;

    const dim3 blk(256);

    // branch 3 grouped convs -> xc
    k_conv_branch3<<<dim3(1025, 16), blk, 0, stream>>>(
        x, c1w, c1g, c1b, c1m, c1v, c2w, c2g, c2b, c2m, c2v, xc);

    // branch 1 & 2 ZPool in one pass over x
    k_zpool12<<<dim3(2050, 16), blk, 0, stream>>>(x, z1a, z2a);

    // branch 3 ZPool over xc (WMMA mean)
    k_reduce3_part<<<dim3(16, 16), blk, 0, stream>>>(xc, pmax, psum);
    k_reduce3_final<<<dim3(16), blk, 0, stream>>>(pmax, psum, z3a);

    // gate 0: spatial (2050, 32)
    const int HW0 = 2050 * 32;
    const dim3 g0((HW0 + 255) / 256, 32);
    k_dwconv7<<<g0, blk, 0, stream>>>(z1a, z1b, dww, dwg, dwb, dwm, dwv, 0, 2050, 32);
    k_dwconv7<<<g0, blk, 0, stream>>>(z1b, z1a, dww, dwg, dwb, dwm, dwv, 0, 2050, 32);
    k_pwconv<<<dim3((unsigned)(((size_t)16 * HW0 + 255) / 256)), blk, 0, stream>>>(
        z1a, s0, pww, pwg, pwb, pwm, pwv, 0, HW0);

    // gate 1: spatial (32, 2050)
    k_dwconv7<<<g0, blk, 0, stream>>>(z2a, z2b, dww, dwg, dwb, dwm, dwv, 1, 32, 2050);
    k_dwconv7<<<g0, blk, 0, stream>>>(z2b, z2a, dww, dwg, dwb, dwm, dwv, 1, 32, 2050);
    k_pwconv<<<dim3((unsigned)(((size_t)16 * HW0 + 255) / 256)), blk, 0, stream>>>(
        z2a, s1, pww, pwg, pwb, pwm, pwv, 1, HW0);

    // gate 2: spatial (32, 32)
    const dim3 g2((1024 + 255) / 256, 32);
    k_dwconv7<<<g2, blk, 0, stream>>>(z3a, z3b, dww, dwg, dwb, dwm, dwv, 2, 32, 32);
    k_dwconv7<<<g2, blk, 0, stream>>>(z3b, z3a, dww, dwg, dwb, dwm, dwv, 2, 32, 32);
    k_pwconv<<<dim3((unsigned)(((size_t)16 * 1024 + 255) / 256)), blk, 0, stream>>>(
        z3a, s2, pww, pwg, pwb, pwm, pwv, 2, 1024);

    // final combine
    const size_t total = (size_t)16 * 2050 * 1024;
    k_final<<<dim3((unsigned)((total + 255) / 256)), blk, 0, stream>>>(x, xc, s0, s1, s2, out);
}